// InvariantPointAttention_freROPE_74191265071612
// MI455X (gfx1250) — compile-verified
//
#include <hip/hip_runtime.h>
#include <hip/hip_bf16.h>
#include <math.h>

// ---------------------------------------------------------------------------
// Problem constants (from reference)
// ---------------------------------------------------------------------------
static constexpr int Nn   = 512;   // sequence length
static constexpr int CSd  = 384;   // s channels
static constexpr int CZd  = 128;   // z channels
static constexpr int Hh   = 8;     // heads
static constexpr int GDg  = 16;    // groups
static constexpr int CHd  = 48;    // GD*3
static constexpr int NFf  = 4;     // freqs
static constexpr int PQp  = 8;
static constexpr int PVp  = 12;
static constexpr int Pp   = 20;    // PQ+PV
static constexpr int CZ4d = 32;
static constexpr int ROTD = 192;   // G*NF*3  (contraction dim of scores)
static constexpr int ZLD  = 40;    // fused z-proj cols: 8 bias + 32 pair
static constexpr int FEAT = 1024;  // H*(CZ4+CH+PV*4)
static constexpr int ZROWS_PER_BLK = 64;   // z rows staged per TDM transfer

typedef __attribute__((ext_vector_type(2))) float v2f;
typedef __attribute__((ext_vector_type(8))) float v8f;
typedef __attribute__((ext_vector_type(4))) unsigned int u32x4;
typedef __attribute__((ext_vector_type(8))) int i32x8;
typedef __attribute__((ext_vector_type(4))) int i32x4;

// V_WMMA_F32_16X16X4_F32 : D(16x16 f32) = A(16x4) * B(4x16) + C, full fp32.
__device__ __forceinline__ v8f wmma4(v2f a, v2f b, v8f c) {
  return __builtin_amdgcn_wmma_f32_16x16x4_f32(false, a, false, b, (short)0, c,
                                               false, false);
}

// ---------------------------------------------------------------------------
// Generic fp32 WMMA GEMM:  C(MxN) = A(MxK) * B(KxN) + bias   (row-major all)
// One wave per 16x16 output tile. M,N multiples of 16.
// ---------------------------------------------------------------------------
__global__ void gemm_wmma(const float* __restrict__ A, int lda,
                          const float* __restrict__ B, int ldb,
                          const float* __restrict__ bias,
                          float* __restrict__ C, int ldc,
                          int M, int N, int K) {
  const int lane   = threadIdx.x & 31;
  const int wid    = (blockIdx.x * blockDim.x + threadIdx.x) >> 5;
  const int ntiles = N >> 4;
  const int mt     = wid / ntiles;
  const int nt     = wid - mt * ntiles;
  if (mt * 16 >= M) return;                 // wave-uniform
  const int mr = mt * 16 + (lane & 15);
  const int nc = nt * 16 + (lane & 15);
  const int kh = (lane >> 4) << 1;          // K-half offset (0 or 2)
  v8f acc = {};
  const float* Arow = A + (size_t)mr * lda;
  for (int k = 0; k < K; k += 4) {
    v2f a, b;
    a[0] = Arow[k + kh];
    a[1] = Arow[k + kh + 1];
    b[0] = B[(size_t)(k + kh) * ldb + nc];
    b[1] = B[(size_t)(k + kh + 1) * ldb + nc];
    acc = wmma4(a, b, acc);
  }
  const float bv = bias ? bias[nc] : 0.0f;
  const int r0 = mt * 16 + ((lane >> 4) << 3);
#pragma unroll
  for (int v = 0; v < 8; ++v)
    C[(size_t)(r0 + v) * ldc + nc] = acc[v] + bv;
}

// ---------------------------------------------------------------------------
// Fused z-projection: out(N*N, 40) = z(N*N,128) @ [wb | wdz] + [bb | bdz]
// z (134 MB) is streamed exactly once, moved by the Tensor Data Mover:
// each block TDM-loads a 64x128 fp32 tile of z into LDS (32 KB), waits on
// TENSORcnt, then 4 waves run WMMA against the LDS-resident weights (24 KB).
// ---------------------------------------------------------------------------
__global__ void zproj_kernel(const float* __restrict__ z,
                             const float* __restrict__ wb,
                             const float* __restrict__ bb,
                             const float* __restrict__ wdz,
                             const float* __restrict__ bdz,
                             float* __restrict__ out) {
  __shared__ float W[CZd][48];                 // padded [wb | wdz | 0]
  __shared__ float Zt[ZROWS_PER_BLK][CZd];     // TDM destination tile
  for (int idx = threadIdx.x; idx < CZd * 48; idx += blockDim.x) {
    const int kk = idx / 48, c = idx - kk * 48;
    float w = 0.0f;
    if (c < 8)        w = wb[kk * 8 + c];
    else if (c < 40)  w = wdz[kk * 32 + (c - 8)];
    W[kk][c] = w;
  }

  // --- TDM: stage 64 rows of z into LDS (one descriptor, wave 0 only) ---
  if (threadIdx.x < 32) {
    const unsigned long long ga =
        (unsigned long long)(uintptr_t)(z + (size_t)blockIdx.x *
                                                ZROWS_PER_BLK * CZd);
    const unsigned int la = (unsigned int)(uintptr_t)(&Zt[0][0]);
    // D# group 0: count=1 | lds_addr | global_addr(57b) | type=2
    u32x4 g0 = {1u, la, (unsigned int)(ga & 0xFFFFFFFFu),
                (unsigned int)(((ga >> 32) & 0x01FFFFFFu) | 0x80000000u)};
    // D# group 1: data_size=4B; tensor_dim0=128; tensor_dim1=N*N;
    //             tile_dim0=128, tile_dim1=64; tensor_dim0_stride=128
    i32x8 g1 = {0x00020000,                     // workgroup_mask=0, 4B elems
                (int)(CZd << 16),               // tensor_dim0 lo16
                (int)(((Nn * Nn) & 0xFFFF) << 16),   // dim0 hi | dim1 lo16
                (int)(((Nn * Nn) >> 16) | (CZd << 16)), // dim1 hi | tile_dim0
                ZROWS_PER_BLK,                  // tile_dim1 | tile_dim2=0
                CZd,                            // dim0_stride lo32
                0, 0};
    i32x4 gz4 = {0, 0, 0, 0};                   // groups 2/3 unused (2-D)
    i32x8 gz8 = {0, 0, 0, 0, 0, 0, 0, 0};
    __builtin_amdgcn_tensor_load_to_lds(g0, g1, gz4, gz4, gz8, 0);
    __builtin_amdgcn_s_wait_tensorcnt(0);
  }
  __syncthreads();

  const int lane = threadIdx.x & 31;
  const int wv   = threadIdx.x >> 5;            // 0..3: one 16-row tile each
  const int mrow = lane & 15;
  const int kh   = (lane >> 4) << 1;
  const int nc   = mrow;
  const float* Zr = &Zt[wv * 16 + mrow][0];
  v8f acc0 = {}, acc1 = {}, acc2 = {};
  for (int k = 0; k < CZd; k += 4) {
    v2f a;
    a[0] = Zr[k + kh];
    a[1] = Zr[k + kh + 1];
    v2f b0, b1, b2;
    b0[0] = W[k + kh][nc];          b0[1] = W[k + kh + 1][nc];
    b1[0] = W[k + kh][16 + nc];     b1[1] = W[k + kh + 1][16 + nc];
    b2[0] = W[k + kh][32 + nc];     b2[1] = W[k + kh + 1][32 + nc];
    acc0 = wmma4(a, b0, acc0);
    acc1 = wmma4(a, b1, acc1);
    acc2 = wmma4(a, b2, acc2);
  }
  const size_t r0 = (size_t)blockIdx.x * ZROWS_PER_BLK + wv * 16 +
                    ((lane >> 4) << 3);
  const float bia0 = (nc < 8) ? bb[nc] : bdz[nc - 8];
#pragma unroll
  for (int v = 0; v < 8; ++v) {
    const size_t ro = (r0 + v) * (size_t)ZLD;
    out[ro + nc]      = acc0[v] + bia0;
    out[ro + 16 + nc] = acc1[v] + bdz[8 + nc];
    if (nc < 8) out[ro + 32 + nc] = acc2[v] + bdz[24 + nc];
  }
}

// ---------------------------------------------------------------------------
// Rotation kernel: per (n,h,g) build R_f (Rodrigues, 4 freqs), rotate q/k
// groups, apply freq-softmax * group-softplus * 1/sqrt(3) to Q side.
// Output layout per head: [h][n][g*12 + f*3 + c]  (contraction dim = 192)
// ---------------------------------------------------------------------------
__global__ void rope_rotate_kernel(const float* __restrict__ rotvec,
                                   const float* __restrict__ lfa,
                                   const float* __restrict__ gwv,
                                   const float* __restrict__ fwv,
                                   const float* __restrict__ qbuf,
                                   const float* __restrict__ kvbuf,
                                   float* __restrict__ Qw,
                                   float* __restrict__ Kr) {
  const int tid = blockIdx.x * blockDim.x + threadIdx.x;
  if (tid >= Nn * Hh * GDg) return;
  const int n  = tid >> 7;
  const int hg = tid & 127;
  const int h  = hg >> 4;
  const int g  = hg & 15;

  const float rx = rotvec[n * 3 + 0], ry = rotvec[n * 3 + 1],
              rz = rotvec[n * 3 + 2];
  const float nrm   = sqrtf(rx * rx + ry * ry + rz * rz);
  const float angle = nrm + 1e-8f;
  const float inv   = 1.0f / fmaxf(nrm, 1e-12f);
  const float ax = rx * inv, ay = ry * inv, az = rz * inv;

  // K2 = K@K for K = hat(axis)
  const float K2_00 = -(az * az + ay * ay);
  const float K2_11 = -(az * az + ax * ax);
  const float K2_22 = -(ay * ay + ax * ax);
  const float K2_01 = ax * ay, K2_02 = ax * az, K2_12 = ay * az;

  // softmax(freq_weights)
  const float f0 = fwv[0], f1 = fwv[1], f2 = fwv[2], f3 = fwv[3];
  const float fm = fmaxf(fmaxf(f0, f1), fmaxf(f2, f3));
  const float e0 = __expf(f0 - fm), e1 = __expf(f1 - fm),
              e2 = __expf(f2 - fm), e3 = __expf(f3 - fm);
  const float ei = 1.0f / (e0 + e1 + e2 + e3);
  const float fwn[4] = {e0 * ei, e1 * ei, e2 * ei, e3 * ei};

  // softplus(group_weights[g]) * sqrt(1/16)
  const float gw  = log1pf(__expf(gwv[g])) * 0.25f;
  const float RS3 = 0.57735026918962576f;  // 1/sqrt(3) folded into Q side

  // base_frequencies(): min(8^(-i/2), 1)
  const float fsb[4] = {1.0f, 0.35355339059327379f, 0.125f,
                        0.044194173824159223f};

  const float qx = qbuf[n * CSd + h * CHd + g * 3 + 0];
  const float qy = qbuf[n * CSd + h * CHd + g * 3 + 1];
  const float qz = qbuf[n * CSd + h * CHd + g * 3 + 2];
  const float kx = kvbuf[n * 2 * CSd + h * 96 + g * 3 + 0];
  const float ky = kvbuf[n * 2 * CSd + h * 96 + g * 3 + 1];
  const float kz = kvbuf[n * 2 * CSd + h * 96 + g * 3 + 2];

  float* Qo = Qw + ((size_t)h * Nn + n) * ROTD + g * 12;
  float* Ko = Kr + ((size_t)h * Nn + n) * ROTD + g * 12;

#pragma unroll
  for (int f = 0; f < NFf; ++f) {
    const float freq = fsb[f] * __expf(lfa[f]);
    const float th = angle * freq;
    const float st = __sinf(th), ct = __cosf(th);
    const float oc = 1.0f - ct;
    const float R00 = 1.0f + oc * K2_00;
    const float R01 = -st * az + oc * K2_01;
    const float R02 =  st * ay + oc * K2_02;
    const float R10 =  st * az + oc * K2_01;
    const float R11 = 1.0f + oc * K2_11;
    const float R12 = -st * ax + oc * K2_12;
    const float R20 = -st * ay + oc * K2_02;
    const float R21 =  st * ax + oc * K2_12;
    const float R22 = 1.0f + oc * K2_22;

    const float sc = fwn[f] * gw * RS3;
    Qo[f * 3 + 0] = (R00 * qx + R01 * qy + R02 * qz) * sc;
    Qo[f * 3 + 1] = (R10 * qx + R11 * qy + R12 * qz) * sc;
    Qo[f * 3 + 2] = (R20 * qx + R21 * qy + R22 * qz) * sc;
    Ko[f * 3 + 0] = (R00 * kx + R01 * ky + R02 * kz);
    Ko[f * 3 + 1] = (R10 * kx + R11 * ky + R12 * kz);
    Ko[f * 3 + 2] = (R20 * kx + R21 * ky + R22 * kz);
  }
}

// ---------------------------------------------------------------------------
// Scores + softmax: one block per (head, 16-row tile).
// Waves compute 16x512 score strip via WMMA (Q' * K'^T, K=192), add bias and
// mask into LDS, then a 2-pass softmax writes a[h][i][j] to global.
// ---------------------------------------------------------------------------
__global__ void scores_softmax_kernel(const float* __restrict__ Qw,
                                      const float* __restrict__ Kr,
                                      const float* __restrict__ zproj,
                                      const float* __restrict__ mask,
                                      float* __restrict__ abuf) {
  __shared__ float S[16][Nn];          // 32 KB
  __shared__ float red[16][16];
  const int h  = blockIdx.x >> 5;
  const int it = blockIdx.x & 31;
  const int wv   = threadIdx.x >> 5;
  const int lane = threadIdx.x & 31;
  const int mrow = lane & 15;
  const int kh   = (lane >> 4) << 1;
  const float* Qh = Qw + ((size_t)h * Nn + it * 16) * ROTD;
  const float* Kh = Kr + (size_t)h * Nn * ROTD;
  const float RS3 = 0.57735026918962576f;  // sqrt(1/3) on the bias term

  for (int jt = wv; jt < 32; jt += 8) {
    const int jc = jt * 16 + mrow;
    v8f acc = {};
    const float* Arow = Qh + mrow * ROTD;
    const float* Brow = Kh + (size_t)jc * ROTD;   // column jc of K'^T
    for (int k = 0; k < ROTD; k += 4) {
      v2f a, b;
      a[0] = Arow[k + kh]; a[1] = Arow[k + kh + 1];
      b[0] = Brow[k + kh]; b[1] = Brow[k + kh + 1];
      acc = wmma4(a, b, acc);
    }
#pragma unroll
    for (int v = 0; v < 8; ++v) {
      const int rloc = v + ((lane >> 4) << 3);
      const int i = it * 16 + rloc;
      const int j = jt * 16 + mrow;
      const float bias = RS3 * zproj[((size_t)i * Nn + j) * ZLD + h];
      const float msk  = 100000.0f * (mask[i] * mask[j] - 1.0f);
      S[rloc][j] = acc[v] + bias + msk;
    }
  }
  __syncthreads();

  // softmax: 16 threads per row
  const int r = threadIdx.x >> 4;
  const int c = threadIdx.x & 15;
  float mx = -1e30f;
  for (int j = c; j < Nn; j += 16) mx = fmaxf(mx, S[r][j]);
  red[r][c] = mx;
  __syncthreads();
  float rowmax = -1e30f;
#pragma unroll
  for (int t = 0; t < 16; ++t) rowmax = fmaxf(rowmax, red[r][t]);
  float sum = 0.0f;
  for (int j = c; j < Nn; j += 16) {
    const float e = __expf(S[r][j] - rowmax);
    S[r][j] = e;
    sum += e;
  }
  __syncthreads();
  red[r][c] = sum;
  __syncthreads();
  float rowsum = 0.0f;
#pragma unroll
  for (int t = 0; t < 16; ++t) rowsum += red[r][t];
  const float invs = 1.0f / rowsum;
  float* arow = abuf + (size_t)h * Nn * Nn + (size_t)(it * 16 + r) * Nn;
  for (int j = c; j < Nn; j += 16) arow[j] = S[r][j] * invs;
}

// ---------------------------------------------------------------------------
// o / o_pt / o_pt_norm: one wave per (head, 16-row tile). B is the gathered
// [v(48) | v_pts(36) | pad(12)] matrix; 16x96 result staged in LDS for the
// cross-lane point-norm, then scattered into the feats buffer.
// ---------------------------------------------------------------------------
__device__ __forceinline__ float load_vcat(const float* __restrict__ kvbuf,
                                           const float* __restrict__ kvpbuf,
                                           int j, int h, int col) {
  if (col < 48) return kvbuf[(size_t)j * (2 * CSd) + h * 96 + 48 + col];
  if (col < 84) {
    const int q = col - 48;
    const int p = q / 3, c = q - p * 3;
    return kvpbuf[(size_t)j * 480 + c * 160 + h * Pp + PQp + p];
  }
  return 0.0f;
}

__global__ void attn_out_kernel(const float* __restrict__ abuf,
                                const float* __restrict__ kvbuf,
                                const float* __restrict__ kvpbuf,
                                float* __restrict__ feats) {
  __shared__ float T[8][16][96];       // 48 KB
  const int lane = threadIdx.x & 31;
  const int wv   = threadIdx.x >> 5;
  const int wid  = blockIdx.x * 8 + wv;
  const int h    = wid >> 5;
  const int it   = wid & 31;
  const int mrow = lane & 15;
  const int kh   = (lane >> 4) << 1;

  v8f acc[6] = {};
  const float* Ah = abuf + (size_t)h * Nn * Nn + (size_t)(it * 16 + mrow) * Nn;
  for (int k = 0; k < Nn; k += 4) {
    v2f a;
    a[0] = Ah[k + kh];
    a[1] = Ah[k + kh + 1];
#pragma unroll
    for (int t = 0; t < 6; ++t) {
      const int col = t * 16 + mrow;
      v2f b;
      b[0] = load_vcat(kvbuf, kvpbuf, k + kh, h, col);
      b[1] = load_vcat(kvbuf, kvpbuf, k + kh + 1, h, col);
      acc[t] = wmma4(a, b, acc[t]);
    }
  }
  const int rl = (lane >> 4) << 3;
#pragma unroll
  for (int t = 0; t < 6; ++t)
#pragma unroll
    for (int v = 0; v < 8; ++v) T[wv][rl + v][t * 16 + mrow] = acc[t][v];

  // same-wave LDS readback (DS ops are in-order per wave; compiler waits)
  const int r = lane & 15;
  const int i = it * 16 + r;
  float* F = feats + (size_t)i * FEAT;
  if (lane < 16) {
    for (int c = 0; c < CHd; ++c) F[h * CHd + c] = T[wv][r][c];
  } else {
    for (int p = 0; p < PVp; ++p) {
      const float x = T[wv][r][48 + 3 * p];
      const float y = T[wv][r][48 + 3 * p + 1];
      const float zc = T[wv][r][48 + 3 * p + 2];
      F[384 + h * PVp + p] = x;
      F[480 + h * PVp + p] = y;
      F[576 + h * PVp + p] = zc;
      F[672 + h * PVp + p] = sqrtf(x * x + y * y + zc * zc + 1e-8f);
    }
  }
}

// ---------------------------------------------------------------------------
// o_pair: for each i, (8x512) attention rows  @  pair_z[i] (512x32).
// One wave per i; A rows 8..15 padded to zero (predicated loads).
// ---------------------------------------------------------------------------
__global__ void pair_out_kernel(const float* __restrict__ abuf,
                                const float* __restrict__ zproj,
                                float* __restrict__ feats) {
  const int lane = threadIdx.x & 31;
  const int i    = blockIdx.x * 8 + (threadIdx.x >> 5);
  const int mrow = lane & 15;
  const int kh   = (lane >> 4) << 1;
  const bool valid = (mrow < Hh);
  const float* Am = abuf + (size_t)mrow * Nn * Nn + (size_t)i * Nn;
  const float* Zp = zproj + (size_t)i * Nn * ZLD + 8;
  v8f acc0 = {}, acc1 = {};
  for (int k = 0; k < Nn; k += 4) {
    v2f a, b0, b1;
    a[0] = valid ? Am[k + kh] : 0.0f;
    a[1] = valid ? Am[k + kh + 1] : 0.0f;
    b0[0] = Zp[(size_t)(k + kh) * ZLD + mrow];
    b0[1] = Zp[(size_t)(k + kh + 1) * ZLD + mrow];
    b1[0] = Zp[(size_t)(k + kh) * ZLD + 16 + mrow];
    b1[1] = Zp[(size_t)(k + kh + 1) * ZLD + 16 + mrow];
    acc0 = wmma4(a, b0, acc0);
    acc1 = wmma4(a, b1, acc1);
  }
  if (lane < 16) {  // rows (heads) 0..7 live in VGPRs 0..7 of lanes 0..15
    float* F = feats + (size_t)i * FEAT + 768;
#pragma unroll
    for (int v = 0; v < Hh; ++v) {
      F[v * CZ4d + mrow]      = acc0[v];
      F[v * CZ4d + 16 + mrow] = acc1[v];
    }
  }
}

// ---------------------------------------------------------------------------
// Host-side orchestration
// ---------------------------------------------------------------------------
extern "C" void kernel_launch(void* const* d_in, const int* in_sizes, int n_in,
                              void* d_out, int out_size, void* d_ws,
                              size_t ws_size, hipStream_t stream) {
  const float* s      = (const float*)d_in[0];
  const float* z      = (const float*)d_in[1];
  const float* rotvec = (const float*)d_in[2];
  const float* mask   = (const float*)d_in[3];
  const float* wq     = (const float*)d_in[4];
  const float* bq     = (const float*)d_in[5];
  const float* wkv    = (const float*)d_in[6];
  const float* bkv    = (const float*)d_in[7];
  const float* wkvp   = (const float*)d_in[8];
  const float* bkvp   = (const float*)d_in[9];
  const float* wb     = (const float*)d_in[10];
  const float* bb     = (const float*)d_in[11];
  const float* wdz    = (const float*)d_in[12];
  const float* bdz    = (const float*)d_in[13];
  const float* wout   = (const float*)d_in[14];
  const float* bout   = (const float*)d_in[15];
  const float* lfa    = (const float*)d_in[16];
  const float* gwv    = (const float*)d_in[17];
  const float* fwv    = (const float*)d_in[18];
  float* out = (float*)d_out;

  // workspace carve-up (floats); total ~15.5M floats (~62 MB)
  float* ws      = (float*)d_ws;
  float* qbuf    = ws;                       // 512*384
  float* kvbuf   = qbuf + 512 * 384;         // 512*768
  float* kvpbuf  = kvbuf + 512 * 768;        // 512*480
  float* zproj   = kvpbuf + 512 * 480;       // 512*512*40
  float* Qw      = zproj + (size_t)512 * 512 * 40;   // 8*512*192
  float* Kr      = Qw + 8 * 512 * 192;               // 8*512*192
  float* abuf    = Kr + 8 * 512 * 192;               // 8*512*512
  float* feats   = abuf + (size_t)8 * 512 * 512;     // 512*1024

  // 1) input projections (M=512, K=384)
  gemm_wmma<<<96, 256, 0, stream>>>(s, CSd, wq, 384, bq, qbuf, 384, 512, 384, 384);
  gemm_wmma<<<192, 256, 0, stream>>>(s, CSd, wkv, 768, bkv, kvbuf, 768, 512, 768, 384);
  gemm_wmma<<<120, 256, 0, stream>>>(s, CSd, wkvp, 480, bkvp, kvpbuf, 480, 512, 480, 384);

  // 2) fused z projections (bias + pair), z streamed once via TDM
  zproj_kernel<<<(Nn * Nn) / ZROWS_PER_BLK, 128, 0, stream>>>(z, wb, bb, wdz,
                                                              bdz, zproj);

  // 3) rotations + weighting
  rope_rotate_kernel<<<256, 256, 0, stream>>>(rotvec, lfa, gwv, fwv, qbuf, kvbuf,
                                              Qw, Kr);

  // 4) scores + bias + mask + softmax
  scores_softmax_kernel<<<256, 256, 0, stream>>>(Qw, Kr, zproj, mask, abuf);

  // 5) attention-weighted values / points (+norms), and pair output
  attn_out_kernel<<<32, 256, 0, stream>>>(abuf, kvbuf, kvpbuf, feats);
  pair_out_kernel<<<64, 256, 0, stream>>>(abuf, zproj, feats);

  // 6) final output projection
  gemm_wmma<<<96, 256, 0, stream>>>(feats, FEAT, wout, 384, bout, out, 384, 512,
                                    384, FEAT);
}